// build_model_77111842832412
// MI455X (gfx1250) — compile-verified
//
#include <hip/hip_runtime.h>
#include <math.h>

// Model dims
#define B_    32
#define SEQ_  32
#define H_    1024
#define NH_   16
#define HS_   64
#define NL_   4
#define TH_   8
#define TA_   8
#define HIN_  512
#define AIN_  64

typedef __attribute__((ext_vector_type(2))) float v2f;
typedef __attribute__((ext_vector_type(8))) float v8f;

// ---------------- positional encoding (matches numpy float64 computation) ---
__global__ void k_pe_init(float* __restrict__ pe) {
  int idx = blockIdx.x * blockDim.x + threadIdx.x;   // 32*1024 total
  int pos = idx >> 10;
  int i   = idx & 1023;
  double angle = (double)pos / ::pow(10000.0, (double)(2 * (i / 2)) / 1024.0);
  pe[idx] = (float)(((i & 1) == 0) ? ::sin(angle) : ::cos(angle));
}

__global__ void k_zero(float* __restrict__ p, int n) {
  int idx = blockIdx.x * blockDim.x + threadIdx.x;
  if (idx < n) p[idx] = 0.f;
}

// ---------------- fp32 WMMA GEMM: C[M,N] = opt_relu(A[M,K] @ W[K,N]) --------
// One wave (32 lanes) computes a 32x64 output tile via V_WMMA_F32_16X16X4_F32.
// Two A fragments share the four B fragments -> 8 WMMAs per 12 loads per
// K-step (vs 4:10 for a 16x64 tile), halving L2 B-traffic per FLOP.
// grid = (N/64, M/32), block = 32.  Requires M%32==0, N%64==0, K%4==0.
// Lane layout (ISA 7.12.2): A frag: m=lane&15, k=2*(lane>>4)+{0,1}
//                           B frag: n=lane&15, k=2*(lane>>4)+{0,1}
//                           D:      m=r+8*(lane>>4), n=lane&15  (r = vgpr 0..7)
__global__ void k_gemm(const float* __restrict__ A, int lda,
                       const float* __restrict__ W, int ldw,
                       float* __restrict__ C, int ldc,
                       int K, int do_relu) {
  const int lane = threadIdx.x;
  const int lo = lane & 15;
  const int hi = lane >> 4;
  const int row0 = blockIdx.y << 5;
  const int col0 = blockIdx.x << 6;

  v8f acc00 = {}, acc01 = {}, acc02 = {}, acc03 = {};
  v8f acc10 = {}, acc11 = {}, acc12 = {}, acc13 = {};

  const float* Arow0 = A + (size_t)(row0 + lo) * lda + 2 * hi;        // 8B aligned
  const float* Arow1 = A + (size_t)(row0 + 16 + lo) * lda + 2 * hi;
  const float* Wbase = W + (size_t)(2 * hi) * ldw + col0 + lo;

  for (int k0 = 0; k0 < K; k0 += 4) {
    v2f a0 = *(const v2f*)(Arow0 + k0);
    v2f a1 = *(const v2f*)(Arow1 + k0);
    const float* w0 = Wbase + (size_t)k0 * ldw;   // row k0+2*hi
    const float* w1 = w0 + ldw;                   // row k0+2*hi+1
    v2f b0, b1, b2, b3;
    b0.x = w0[0];  b0.y = w1[0];
    b1.x = w0[16]; b1.y = w1[16];
    b2.x = w0[32]; b2.y = w1[32];
    b3.x = w0[48]; b3.y = w1[48];
    acc00 = __builtin_amdgcn_wmma_f32_16x16x4_f32(false, a0, false, b0, (short)0, acc00, false, false);
    acc10 = __builtin_amdgcn_wmma_f32_16x16x4_f32(false, a1, false, b0, (short)0, acc10, false, false);
    acc01 = __builtin_amdgcn_wmma_f32_16x16x4_f32(false, a0, false, b1, (short)0, acc01, false, false);
    acc11 = __builtin_amdgcn_wmma_f32_16x16x4_f32(false, a1, false, b1, (short)0, acc11, false, false);
    acc02 = __builtin_amdgcn_wmma_f32_16x16x4_f32(false, a0, false, b2, (short)0, acc02, false, false);
    acc12 = __builtin_amdgcn_wmma_f32_16x16x4_f32(false, a1, false, b2, (short)0, acc12, false, false);
    acc03 = __builtin_amdgcn_wmma_f32_16x16x4_f32(false, a0, false, b3, (short)0, acc03, false, false);
    acc13 = __builtin_amdgcn_wmma_f32_16x16x4_f32(false, a1, false, b3, (short)0, acc13, false, false);
  }

  float* Crow0 = C + (size_t)(row0 + 8 * hi) * ldc + col0 + lo;
  float* Crow1 = C + (size_t)(row0 + 16 + 8 * hi) * ldc + col0 + lo;
#pragma unroll
  for (int r = 0; r < 8; r++) {
    float u0 = acc00[r], u1 = acc01[r], u2 = acc02[r], u3 = acc03[r];
    float w0v = acc10[r], w1v = acc11[r], w2v = acc12[r], w3v = acc13[r];
    if (do_relu) {
      u0 = fmaxf(u0, 0.f); u1 = fmaxf(u1, 0.f); u2 = fmaxf(u2, 0.f); u3 = fmaxf(u3, 0.f);
      w0v = fmaxf(w0v, 0.f); w1v = fmaxf(w1v, 0.f); w2v = fmaxf(w2v, 0.f); w3v = fmaxf(w3v, 0.f);
    }
    float* cp0 = Crow0 + (size_t)r * ldc;
    float* cp1 = Crow1 + (size_t)r * ldc;
    cp0[0] = u0;  cp0[16] = u1;  cp0[32] = u2;  cp0[48] = u3;
    cp1[0] = w0v; cp1[16] = w1v; cp1[32] = w2v; cp1[48] = w3v;
  }
}

// ---------------- h = tokens + pe * (t < L) ---------------------------------
__global__ void k_add_pe(const float* __restrict__ tok, const float* __restrict__ pe,
                         float* __restrict__ h, int L) {
  int idx = blockIdx.x * blockDim.x + threadIdx.x;   // B*SEQ*H total
  int t = (idx >> 10) & 31;
  float p = (t < L) ? pe[(t << 10) + (idx & 1023)] : 0.f;
  h[idx] = tok[idx] + p;
}

// ---------------- attention core (per b,head; ~1.2% of FLOPs) ---------------
// scores = Q K^T / 8 + m1 ; softmax ; * m2 ; @ V
__global__ void k_attn(const float* __restrict__ Q, const float* __restrict__ K,
                       const float* __restrict__ V, float* __restrict__ O, int L) {
  __shared__ float sQ[SEQ_][HS_];
  __shared__ float sK[SEQ_][HS_];
  __shared__ float sV[SEQ_][HS_];
  __shared__ float sS[SEQ_][SEQ_];
  const int b = blockIdx.x >> 4;
  const int hd = blockIdx.x & 15;
  const size_t base = (size_t)b * SEQ_ * H_ + hd * HS_;

  for (int idx = threadIdx.x; idx < SEQ_ * HS_; idx += blockDim.x) {
    int t = idx >> 6, d = idx & 63;
    size_t g = base + (size_t)t * H_ + d;
    sQ[t][d] = Q[g]; sK[t][d] = K[g]; sV[t][d] = V[g];
  }
  __syncthreads();

  for (int idx = threadIdx.x; idx < SEQ_ * SEQ_; idx += blockDim.x) {
    int q = idx >> 5, k = idx & 31;
    float s = 0.f;
#pragma unroll
    for (int d = 0; d < HS_; d++) s += sQ[q][d] * sK[k][d];
    s *= 0.125f;
    if (!((q < L) && (k < L))) s = -1e20f;
    sS[q][k] = s;
  }
  __syncthreads();

  if (threadIdx.x < SEQ_) {
    int q = threadIdx.x;
    float mx = sS[q][0];
#pragma unroll
    for (int k = 1; k < SEQ_; k++) mx = fmaxf(mx, sS[q][k]);
    float sum = 0.f;
#pragma unroll
    for (int k = 0; k < SEQ_; k++) { float e = __expf(sS[q][k] - mx); sS[q][k] = e; sum += e; }
    float inv = 1.f / sum;
    bool vq = q < L;
#pragma unroll
    for (int k = 0; k < SEQ_; k++) sS[q][k] = (vq && (k < L)) ? sS[q][k] * inv : 0.f;
  }
  __syncthreads();

  for (int idx = threadIdx.x; idx < SEQ_ * HS_; idx += blockDim.x) {
    int q = idx >> 6, d = idx & 63;
    float o = 0.f;
#pragma unroll
    for (int k = 0; k < SEQ_; k++) o += sS[q][k] * sV[k][d];
    O[base + (size_t)q * H_ + d] = o;
  }
}

// ---------------- h = LN(h + delta)*g + b ; optionally forked into out2 -----
__global__ void k_ln(float* __restrict__ h, const float* __restrict__ delta,
                     const float* __restrict__ g, const float* __restrict__ bt,
                     float* __restrict__ out2) {
  __shared__ float red[256];
  const int row = blockIdx.x;
  float* hr = h + (size_t)row * H_;
  const float* dr = delta + (size_t)row * H_;
  float x[4];
  float s = 0.f;
#pragma unroll
  for (int j = 0; j < 4; j++) {
    int c = threadIdx.x + j * 256;
    x[j] = hr[c] + dr[c];
    s += x[j];
  }
  red[threadIdx.x] = s; __syncthreads();
  for (int off = 128; off > 0; off >>= 1) {
    if (threadIdx.x < off) red[threadIdx.x] += red[threadIdx.x + off];
    __syncthreads();
  }
  float mean = red[0] * (1.f / H_);
  __syncthreads();
  float v = 0.f;
#pragma unroll
  for (int j = 0; j < 4; j++) { float d = x[j] - mean; v += d * d; }
  red[threadIdx.x] = v; __syncthreads();
  for (int off = 128; off > 0; off >>= 1) {
    if (threadIdx.x < off) red[threadIdx.x] += red[threadIdx.x + off];
    __syncthreads();
  }
  float inv = rsqrtf(red[0] * (1.f / H_) + 1e-5f);
#pragma unroll
  for (int j = 0; j < 4; j++) {
    int c = threadIdx.x + j * 256;
    float val = (x[j] - mean) * inv * g[c] + bt[c];
    hr[c] = val;
    if (out2) out2[(size_t)row * H_ + c] = val;
  }
}

// ---------------- r = sigmoid(hf @ Wr) -------------------------------------
__global__ void k_dot_sig(const float* __restrict__ h, const float* __restrict__ Wr,
                          float* __restrict__ out, int iter) {
  __shared__ float red[256];
  const int b = blockIdx.x;
  const float* hb = h + (size_t)b * SEQ_ * H_;
  float s = 0.f;
  for (int c = threadIdx.x; c < SEQ_ * H_; c += 256) s += hb[c] * Wr[c];
  red[threadIdx.x] = s; __syncthreads();
  for (int off = 128; off > 0; off >>= 1) {
    if (threadIdx.x < off) red[threadIdx.x] += red[threadIdx.x + off];
    __syncthreads();
  }
  if (threadIdx.x == 0) out[b * TA_ + iter] = 1.f / (1.f + __expf(-red[0]));
}

// ---------------------------------------------------------------------------
extern "C" void kernel_launch(void* const* d_in, const int* in_sizes, int n_in,
                              void* d_out, int out_size, void* d_ws, size_t ws_size,
                              hipStream_t stream) {
  const float* hist_s = (const float*)d_in[0];   // [B,TH,HIN]
  const float* hist_a = (const float*)d_in[1];   // [B,TH,AIN]
  const float* s_in   = (const float*)d_in[2];   // [B,HIN]
  const float* a_list = (const float*)d_in[3];   // [B,TA,AIN]
  const float* Ws  = (const float*)d_in[4];      // [HIN,H]
  const float* Wa  = (const float*)d_in[5];      // [AIN,H]
  const float* Wq  = (const float*)d_in[6];      // [NL,H,H]
  const float* Wk  = (const float*)d_in[7];
  const float* Wv  = (const float*)d_in[8];
  const float* Wo  = (const float*)d_in[9];
  const float* Wfc = (const float*)d_in[10];
  const float* ln1g = (const float*)d_in[11];    // [NL,H]
  const float* ln1b = (const float*)d_in[12];
  const float* ln2g = (const float*)d_in[13];
  const float* ln2b = (const float*)d_in[14];
  const float* Wr  = (const float*)d_in[15];     // [SEQ*H,1]
  const float* Wso = (const float*)d_in[16];     // [SEQ*H,HIN]
  float* out = (float*)d_out;                    // 256 r's then 32*8*512 s's
  float* ws  = (float*)d_ws;

  const int T = SEQ_ * H_;           // 32768 elems per batch row-block
  const int TENS = B_ * T;           // 1<<20 floats per activation tensor
  float* pe  = ws;                   // 32768
  float* tok = pe + SEQ_ * H_;       // tokens  [B,32,H]
  float* h   = tok + TENS;           // working h
  float* Qb  = h   + TENS;
  float* Kb  = Qb  + TENS;
  float* Vb  = Kb  + TENS;
  float* AO  = Vb  + TENS;
  float* TMP = AO  + TENS;
  float* PH  = TMP + TENS;           // prev_h[4]

  auto gemm = [&](const float* A, int lda, const float* W, int ldw,
                  float* C, int ldc, int M, int N, int K, int relu) {
    dim3 grid(N / 64, M / 32);
    k_gemm<<<grid, 32, 0, stream>>>(A, lda, W, ldw, C, ldc, K, relu);
  };

  k_pe_init<<<(SEQ_ * H_) / 256, 256, 0, stream>>>(pe);
  k_zero<<<TENS / 256, 256, 0, stream>>>(tok, TENS);

  // interleave embedded history: tok[:,2t]=relu(hs), tok[:,2t+1]=relu(ha)
  for (int t = 0; t < TH_; t++) {
    gemm(hist_s + t * HIN_, TH_ * HIN_, Ws, H_, tok + (2 * t) * H_,     T, B_, H_, HIN_, 1);
    gemm(hist_a + t * AIN_, TH_ * AIN_, Wa, H_, tok + (2 * t + 1) * H_, T, B_, H_, AIN_, 1);
  }
  gemm(s_in, HIN_, Ws, H_, tok + 16 * H_, T, B_, H_, HIN_, 1);   // tok[:,16]

  for (int i = 0; i < TA_; i++) {
    const int pos_a = 2 * TH_ + 1 + 2 * i;                       // 17+2i
    gemm(a_list + i * AIN_, TA_ * AIN_, Wa, H_, tok + pos_a * H_, T, B_, H_, AIN_, 1);
    const int L = pos_a + 1;
    k_add_pe<<<TENS / 256, 256, 0, stream>>>(tok, pe, h, L);

    for (int j = 0; j < NL_; j++) {
      const float* qk = (i == 0) ? (const float*)h : (const float*)(PH + (size_t)j * TENS);
      gemm(qk, H_, Wq + (size_t)j * H_ * H_, H_, Qb, H_, B_ * SEQ_, H_, H_, 0);
      gemm(qk, H_, Wk + (size_t)j * H_ * H_, H_, Kb, H_, B_ * SEQ_, H_, H_, 0);
      gemm(h,  H_, Wv + (size_t)j * H_ * H_, H_, Vb, H_, B_ * SEQ_, H_, H_, 0);
      k_attn<<<B_ * NH_, 256, 0, stream>>>(Qb, Kb, Vb, AO, L);
      gemm(AO, H_, Wo + (size_t)j * H_ * H_, H_, TMP, H_, B_ * SEQ_, H_, H_, 0);
      k_ln<<<B_ * SEQ_, 256, 0, stream>>>(h, TMP, ln1g + j * H_, ln1b + j * H_, nullptr);
      gemm(h, H_, Wfc + (size_t)j * H_ * H_, H_, TMP, H_, B_ * SEQ_, H_, H_, 0);
      // second LN also forks the layer output into prev_h[j] for iteration i+1
      k_ln<<<B_ * SEQ_, 256, 0, stream>>>(h, TMP, ln2g + j * H_, ln2b + j * H_,
                                          PH + (size_t)j * TENS);
    }

    // r[b,i] = sigmoid(hf @ Wr)
    k_dot_sig<<<B_, 256, 0, stream>>>(h, Wr, out, i);
    // s_new[b,i,:] = relu(hf @ Wso)   -> written straight into d_out layout [B,TA,HIN]
    float* s_new = out + B_ * TA_ + i * HIN_;                    // row b at stride TA_*HIN_
    gemm(h, T, Wso, HIN_, s_new, TA_ * HIN_, B_, HIN_, T, 1);
    // tok[:, pos_a+1] = relu(s_new @ Ws)
    gemm(s_new, TA_ * HIN_, Ws, H_, tok + (pos_a + 1) * H_, T, B_, H_, HIN_, 1);
  }
}